// CausalGraphNet_64776696758632
// MI455X (gfx1250) — compile-verified
//
#include <hip/hip_runtime.h>

#define NNODES 100000
#define NEDGES 1600000
#define DF     64

typedef __attribute__((ext_vector_type(2))) float v2f;
typedef __attribute__((ext_vector_type(8))) float v8f;

// ---------------------------------------------------------------------------
// Utility fills (ws must be re-initialized every call: graph replay safe)
// ---------------------------------------------------------------------------
__global__ void fill_kernel(float* __restrict__ p, float val, int n) {
    int i = blockIdx.x * blockDim.x + threadIdx.x;
    if (i < n) p[i] = val;
}

__global__ void zero4_kernel(float4* __restrict__ p, int n4) {
    int i = blockIdx.x * blockDim.x + threadIdx.x;
    if (i < n4) p[i] = make_float4(0.f, 0.f, 0.f, 0.f);
}

// ---------------------------------------------------------------------------
// Degree (in-degree + self loop) and D^{-1/2}
// ---------------------------------------------------------------------------
__global__ void deg_kernel(const int* __restrict__ ei, float* __restrict__ deg) {
    int e = blockIdx.x * blockDim.x + threadIdx.x;
    if (e < NEDGES) atomicAdd(&deg[ei[NEDGES + e]], 1.0f);
}

__global__ void dinv_kernel(const float* __restrict__ deg, float* __restrict__ dinv) {
    int i = blockIdx.x * blockDim.x + threadIdx.x;
    if (i < NNODES) dinv[i] = rsqrtf(deg[i]);
}

// ---------------------------------------------------------------------------
// H = X @ W  for X:[N,64], W:[64,64] using fp32 WMMA 16x16x4.
// One block = 128 threads = 4 waves; block b owns row tile b, wave w owns
// column tile w. K=64 consumed in 16 steps of K=4.
//
// A (16x4 f32) lane layout: lanes 0-15 -> M=lane, VGPR0/1 = K=0/1;
//                           lanes16-31 -> M=lane-16, VGPR0/1 = K=2/3.
// B (4x16)     lane layout: N on lanes (0-15 / 16-31), K split like A.
// C/D (16x16)  VGPR r: M=r (lanes 0-15), M=8+r (lanes 16-31); N = lane&15.
// ---------------------------------------------------------------------------
__global__ __launch_bounds__(128)
void gemm64_wmma_kernel(const float* __restrict__ X,
                        const float* __restrict__ W,
                        float* __restrict__ H) {
    const int lane  = threadIdx.x & 31;
    const int wave  = threadIdx.x >> 5;
    const int m0    = blockIdx.x * 16;
    const int n0    = wave * 16;
    const int row   = m0 + (lane & 15);
    const int col   = n0 + (lane & 15);
    const int khalf = (lane >> 4) * 2;      // 0 for lanes 0-15, 2 for 16-31

    v8f acc = {};
#pragma unroll
    for (int k0 = 0; k0 < DF; k0 += 4) {
        v2f a, b;
        const float2 av =
            *reinterpret_cast<const float2*>(X + (size_t)row * DF + k0 + khalf);
        a[0] = av.x;
        a[1] = av.y;
        b[0] = W[(size_t)(k0 + khalf)     * DF + col];
        b[1] = W[(size_t)(k0 + khalf + 1) * DF + col];
        acc = __builtin_amdgcn_wmma_f32_16x16x4_f32(
            /*neg_a=*/false, a, /*neg_b=*/false, b,
            /*c_mod=*/(short)0, acc, /*reuse_a=*/false, /*reuse_b=*/false);
    }

    const int rbase = m0 + ((lane >> 4) ? 8 : 0);
#pragma unroll
    for (int r = 0; r < 8; ++r)
        H[(size_t)(rbase + r) * DF + col] = acc[r];
}

// ---------------------------------------------------------------------------
// Edge scatter: agg[dst] += h[src] * dinv[src]*dinv[dst]
// 16 lanes per edge, float4 gather per lane, 4 f32 atomics per lane.
// h and agg are 25.6 MB each -> resident in the 192 MB L2.
// ---------------------------------------------------------------------------
__global__ void edge_agg_kernel(const float* __restrict__ h,
                                const int* __restrict__ ei,
                                const float* __restrict__ dinv,
                                float* __restrict__ agg) {
    int tid = blockIdx.x * blockDim.x + threadIdx.x;   // E*16 = 25.6M < 2^31
    int e = tid >> 4;
    int q = tid & 15;
    if (e >= NEDGES) return;
    int s = ei[e];
    int d = ei[NEDGES + e];
    float nrm = dinv[s] * dinv[d];
    float4 v = reinterpret_cast<const float4*>(h + (size_t)s * DF)[q];
    float* o = agg + (size_t)d * DF + q * 4;
    atomicAdd(o + 0, v.x * nrm);
    atomicAdd(o + 1, v.y * nrm);
    atomicAdd(o + 2, v.z * nrm);
    atomicAdd(o + 3, v.w * nrm);
}

// ---------------------------------------------------------------------------
// Per-layer epilogue: out = agg + h*dinv^2 + b, optional ReLU
// ---------------------------------------------------------------------------
__global__ void post_kernel(const float* __restrict__ agg,
                            const float* __restrict__ h,
                            const float* __restrict__ dinv,
                            const float* __restrict__ b,
                            float* __restrict__ xout,
                            int relu) {
    int i = blockIdx.x * blockDim.x + threadIdx.x;
    if (i >= NNODES * DF) return;
    int node = i >> 6;
    int f    = i & (DF - 1);
    float di = dinv[node];
    float v  = agg[i] + h[i] * di * di + b[f];
    xout[i]  = relu ? fmaxf(v, 0.f) : v;
}

// ---------------------------------------------------------------------------
// Layer 3: 64 -> 1 projection (per-node dot product)
// ---------------------------------------------------------------------------
__global__ void dot64_kernel(const float* __restrict__ x,
                             const float* __restrict__ w3,
                             float* __restrict__ h3) {
    int i = blockIdx.x * blockDim.x + threadIdx.x;
    if (i >= NNODES) return;
    const float4* xp = reinterpret_cast<const float4*>(x + (size_t)i * DF);
    float s = 0.f;
#pragma unroll
    for (int q = 0; q < 16; ++q) {
        float4 v = xp[q];
        s = fmaf(v.x, w3[q * 4 + 0],
            fmaf(v.y, w3[q * 4 + 1],
            fmaf(v.z, w3[q * 4 + 2],
            fmaf(v.w, w3[q * 4 + 3], s))));
    }
    h3[i] = s;
}

__global__ void edge_agg1_kernel(const float* __restrict__ h3,
                                 const int* __restrict__ ei,
                                 const float* __restrict__ dinv,
                                 float* __restrict__ agg3) {
    int e = blockIdx.x * blockDim.x + threadIdx.x;
    if (e >= NEDGES) return;
    int s = ei[e];
    int d = ei[NEDGES + e];
    atomicAdd(&agg3[d], h3[s] * dinv[s] * dinv[d]);
}

__global__ void final_kernel(const float* __restrict__ agg3,
                             const float* __restrict__ h3,
                             const float* __restrict__ dinv,
                             const float* __restrict__ b3,
                             float* __restrict__ out) {
    int i = blockIdx.x * blockDim.x + threadIdx.x;
    if (i >= NNODES) return;
    float di = dinv[i];
    out[i] = agg3[i] + h3[i] * di * di + b3[0];
}

// ---------------------------------------------------------------------------
extern "C" void kernel_launch(void* const* d_in, const int* in_sizes, int n_in,
                              void* d_out, int out_size, void* d_ws, size_t ws_size,
                              hipStream_t stream) {
    const float* x  = (const float*)d_in[0];
    const int*   ei = (const int*)d_in[1];     // [2,E] flat: src = ei[e], dst = ei[E+e]
    const float* W1 = (const float*)d_in[2];
    const float* b1 = (const float*)d_in[3];
    const float* W2 = (const float*)d_in[4];
    const float* b2 = (const float*)d_in[5];
    const float* W3 = (const float*)d_in[6];
    const float* b3 = (const float*)d_in[7];
    float* out = (float*)d_out;

    // Workspace layout (floats): deg[N] dinv[N] h3[N] agg3[N] hbuf[N*64] aggbuf[N*64] xbuf[N*64]
    float* ws     = (float*)d_ws;
    float* deg    = ws;
    float* dinv   = ws + (size_t)NNODES;
    float* h3     = ws + (size_t)2 * NNODES;
    float* agg3   = ws + (size_t)3 * NNODES;
    float* hbuf   = ws + (size_t)4 * NNODES;
    float* aggbuf = hbuf + (size_t)NNODES * DF;
    float* xbuf   = aggbuf + (size_t)NNODES * DF;

    const int B = 256;
    const int gN    = (NNODES + B - 1) / B;
    const int gE    = (NEDGES + B - 1) / B;
    const int gNF   = (NNODES * DF + B - 1) / B;
    const int gNF4  = (NNODES * DF / 4 + B - 1) / B;
    const int gE16  = (NEDGES * 16 + B - 1) / B;
    const int gGemm = NNODES / 16;             // 6250, exact

    // Normalization (shared by all layers)
    fill_kernel<<<gN, B, 0, stream>>>(deg, 1.0f, NNODES);
    deg_kernel<<<gE, B, 0, stream>>>(ei, deg);
    dinv_kernel<<<gN, B, 0, stream>>>(deg, dinv);

    // Layer 1
    gemm64_wmma_kernel<<<gGemm, 128, 0, stream>>>(x, W1, hbuf);
    zero4_kernel<<<gNF4, B, 0, stream>>>((float4*)aggbuf, NNODES * DF / 4);
    edge_agg_kernel<<<gE16, B, 0, stream>>>(hbuf, ei, dinv, aggbuf);
    post_kernel<<<gNF, B, 0, stream>>>(aggbuf, hbuf, dinv, b1, xbuf, 1);

    // Layer 2
    gemm64_wmma_kernel<<<gGemm, 128, 0, stream>>>(xbuf, W2, hbuf);
    zero4_kernel<<<gNF4, B, 0, stream>>>((float4*)aggbuf, NNODES * DF / 4);
    edge_agg_kernel<<<gE16, B, 0, stream>>>(hbuf, ei, dinv, aggbuf);
    post_kernel<<<gNF, B, 0, stream>>>(aggbuf, hbuf, dinv, b2, xbuf, 1);

    // Layer 3 (64 -> 1)
    dot64_kernel<<<gN, B, 0, stream>>>(xbuf, W3, h3);
    fill_kernel<<<gN, B, 0, stream>>>(agg3, 0.0f, NNODES);
    edge_agg1_kernel<<<gE, B, 0, stream>>>(h3, ei, dinv, agg3);
    final_kernel<<<gN, B, 0, stream>>>(agg3, h3, dinv, b3, out);
}